// network_representation_module_GCN_57346403336482
// MI455X (gfx1250) — compile-verified
//
#include <hip/hip_runtime.h>
#include <hip/hip_bf16.h>

// ---------------------------------------------------------------------------
// 2-layer GraphConv (DGL norm='both') + BatchNorm1d + PReLU on gfx1250.
//
// Per layer:   h = scatter_add_{dst}( ((x*norm_src) @ W)[src] ) * norm_dst + b
//              h = PReLU( BN(h) )
//
// GEMM uses V_WMMA_F32_16X16X4_F32 (fp32 in / fp32 acc) so the math matches
// the fp32 reference exactly (modulo accumulation order).
// ---------------------------------------------------------------------------

typedef float v2f __attribute__((ext_vector_type(2)));
typedef float v8f __attribute__((ext_vector_type(8)));

#define FEATS 128
#define BN_EPS 1e-5f

// ---------------------------------------------------------------------------
// generic zero-fill
__global__ void zero_f32_kernel(float* __restrict__ p, long long n) {
    long long i = (long long)blockIdx.x * blockDim.x + threadIdx.x;
    if (i < n) p[i] = 0.0f;
}

// ---------------------------------------------------------------------------
// degree accumulation: deg_out[src]++ , deg_in[dst]++
__global__ void degree_kernel(const long long* __restrict__ edges,
                              float* __restrict__ deg_out,
                              float* __restrict__ deg_in, int nEdges) {
    int e = blockIdx.x * blockDim.x + threadIdx.x;
    if (e >= nEdges) return;
    int s = (int)edges[e];
    int d = (int)edges[nEdges + e];
    atomicAdd(&deg_out[s], 1.0f);
    atomicAdd(&deg_in[d], 1.0f);
}

// in-place: deg -> rsqrt(max(deg, 1))
__global__ void norm_kernel(float* __restrict__ deg, int n) {
    int i = blockIdx.x * blockDim.x + threadIdx.x;
    if (i >= n) return;
    float d = deg[i];
    d = d < 1.0f ? 1.0f : d;
    deg[i] = rsqrtf(d);
}

// ---------------------------------------------------------------------------
// Y[n, :] = (X[n, :] * nrm[n]) @ W      (X: nRows x 128, W: 128 x 128 row-major,
//                                        this block handles output cols n0..n0+15)
// Block: 256 threads = 8 waves; block tile = 128 rows x 16 cols; K tiled by 32.
__global__ __launch_bounds__(256)
void gemm_gcn_wmma(const float* __restrict__ X, const float* __restrict__ nrm,
                   const float* __restrict__ W, float* __restrict__ Y, int nRows) {
    __shared__ float sX[128][33];   // 33 = pad: 33*m mod 64 injective -> no bank conflicts
    __shared__ float sW[32][16];

    const int t    = threadIdx.x;
    const int wave = t >> 5;
    const int lane = t & 31;
    const int lg   = lane >> 4;     // lane group 0/1
    const int lm   = lane & 15;

    const int m0 = blockIdx.x * 128;
    const int n0 = blockIdx.y * 16;

    v8f acc = {0.f, 0.f, 0.f, 0.f, 0.f, 0.f, 0.f, 0.f};

    for (int kt = 0; kt < FEATS; kt += 32) {
        // stage X chunk (128 rows x 32 K) scaled by norm_src, as float4 quads
        for (int i = t; i < 1024; i += 256) {
            int row  = i >> 3;          // 8 quads per row
            int q    = i & 7;
            int grow = m0 + row;
            float4 v = make_float4(0.f, 0.f, 0.f, 0.f);
            if (grow < nRows) {
                const float4 src =
                    *(const float4*)(X + (size_t)grow * FEATS + kt + q * 4);
                float s = nrm[grow];
                v = make_float4(src.x * s, src.y * s, src.z * s, src.w * s);
            }
            sX[row][q * 4 + 0] = v.x;
            sX[row][q * 4 + 1] = v.y;
            sX[row][q * 4 + 2] = v.z;
            sX[row][q * 4 + 3] = v.w;
        }
        // stage W chunk (32 K x 16 cols)
        for (int i = t; i < 512; i += 256) {
            int k = i >> 4;
            int n = i & 15;
            sW[k][n] = W[(size_t)(kt + k) * FEATS + n0 + n];
        }
        __syncthreads();

#pragma unroll
        for (int kk = 0; kk < 32; kk += 4) {
            int k = kk + lg * 2;
            v2f a, b;
            a[0] = sX[wave * 16 + lm][k];
            a[1] = sX[wave * 16 + lm][k + 1];
            b[0] = sW[k][lm];
            b[1] = sW[k + 1][lm];
            // D = A(16x4,f32) x B(4x16,f32) + C  -> v_wmma_f32_16x16x4_f32
            acc = __builtin_amdgcn_wmma_f32_16x16x4_f32(
                false, a, false, b, (short)0, acc, false, false);
        }
        __syncthreads();
    }

#pragma unroll
    for (int r = 0; r < 8; ++r) {
        int row = m0 + wave * 16 + r + 8 * lg;
        if (row < nRows) Y[(size_t)row * FEATS + n0 + lm] = acc[r];
    }
}

// ---------------------------------------------------------------------------
// AGG[dst, :] += H[src, :]   — one thread per (edge, float4-of-features)
__global__ void scatter_kernel(const long long* __restrict__ edges,
                               const float* __restrict__ H,
                               float* __restrict__ AGG, int nEdges) {
    long long idx = (long long)blockIdx.x * blockDim.x + threadIdx.x;
    long long total = (long long)nEdges * 32;   // 128 feats / 4
    if (idx >= total) return;
    int e = (int)(idx >> 5);
    int q = (int)(idx & 31);
    int s = (int)edges[e];
    int d = (int)edges[nEdges + e];
    const float4 hv = *(const float4*)(H + (size_t)s * FEATS + q * 4);
    float* ap = AGG + (size_t)d * FEATS + q * 4;
    atomicAdd(ap + 0, hv.x);
    atomicAdd(ap + 1, hv.y);
    atomicAdd(ap + 2, hv.z);
    atomicAdd(ap + 3, hv.w);
}

// ---------------------------------------------------------------------------
// Accumulate per-channel sum / sumsq of v = AGG*norm_dst + bias into stats[256].
__global__ __launch_bounds__(128)
void bn_stats_kernel(const float* __restrict__ AGG, const float* __restrict__ ndst,
                     const float* __restrict__ bias, float* __restrict__ stats,
                     int nRows) {
    const int c = threadIdx.x;            // channel 0..127
    int r0 = blockIdx.x * 64;
    int r1 = r0 + 64;
    if (r1 > nRows) r1 = nRows;
    float bc = bias[c];
    float s = 0.f, ss = 0.f;
    for (int r = r0; r < r1; ++r) {
        float v = AGG[(size_t)r * FEATS + c] * ndst[r] + bc;
        s += v;
        ss += v * v;
    }
    atomicAdd(&stats[c], s);
    atomicAdd(&stats[FEATS + c], ss);
}

// ---------------------------------------------------------------------------
// OUT = PReLU( BN( AGG*norm_dst + bias ) )     (safe to run in-place, OUT==AGG)
__global__ void bn_apply_kernel(const float* __restrict__ AGG,
                                const float* __restrict__ ndst,
                                const float* __restrict__ bias,
                                const float* __restrict__ stats,
                                const float* __restrict__ gamma,
                                const float* __restrict__ beta,
                                const float* __restrict__ alpha,
                                float* __restrict__ OUT, int nRows) {
    long long idx = (long long)blockIdx.x * blockDim.x + threadIdx.x;
    long long total = (long long)nRows * FEATS;
    if (idx >= total) return;
    int c = (int)(idx & (FEATS - 1));
    int r = (int)(idx >> 7);
    float invN = 1.0f / (float)nRows;
    float mean = stats[c] * invN;
    float var  = stats[FEATS + c] * invN - mean * mean;   // biased, as torch BN
    float rstd = rsqrtf(var + BN_EPS);
    float v = AGG[idx] * ndst[r] + bias[c];
    float y = (v - mean) * rstd * gamma[c] + beta[c];
    OUT[idx] = y >= 0.f ? y : alpha[0] * y;
}

// ---------------------------------------------------------------------------
extern "C" void kernel_launch(void* const* d_in, const int* in_sizes, int n_in,
                              void* d_out, int out_size, void* d_ws, size_t ws_size,
                              hipStream_t stream) {
    const float*     x     = (const float*)d_in[0];
    const float*     W1    = (const float*)d_in[1];
    const float*     b1    = (const float*)d_in[2];
    const float*     g1    = (const float*)d_in[3];
    const float*     be1   = (const float*)d_in[4];
    const float*     a1    = (const float*)d_in[5];
    const float*     W2    = (const float*)d_in[6];
    const float*     b2    = (const float*)d_in[7];
    const float*     g2    = (const float*)d_in[8];
    const float*     be2   = (const float*)d_in[9];
    const float*     a2    = (const float*)d_in[10];
    const long long* edges = (const long long*)d_in[11];   // int64 (2, E)

    const int N = in_sizes[0] / FEATS;     // 50000
    const int E = in_sizes[11] / 2;        // 600000

    // Workspace: B (N*128), norm_src (N), norm_dst (N), stats (256). ~26 MB.
    float* wsf   = (float*)d_ws;
    float* B     = wsf;
    float* nsrc  = wsf + (size_t)N * FEATS;
    float* ndst  = nsrc + N;
    float* stats = ndst + N;
    float* A     = (float*)d_out;          // d_out doubles as scratch buffer A

    const long long NF = (long long)N * FEATS;
    dim3 gemmGrid((N + 127) / 128, FEATS / 16);
    int  scatBlocks = (int)(((long long)E * 32 + 255) / 256);

    // ---- degrees -> norms ------------------------------------------------
    zero_f32_kernel<<<(N + 255) / 256, 256, 0, stream>>>(nsrc, N);
    zero_f32_kernel<<<(N + 255) / 256, 256, 0, stream>>>(ndst, N);
    degree_kernel<<<(E + 255) / 256, 256, 0, stream>>>(edges, nsrc, ndst, E);
    norm_kernel<<<(N + 255) / 256, 256, 0, stream>>>(nsrc, N);
    norm_kernel<<<(N + 255) / 256, 256, 0, stream>>>(ndst, N);

    // ---- layer 1 ---------------------------------------------------------
    gemm_gcn_wmma<<<gemmGrid, 256, 0, stream>>>(x, nsrc, W1, A, N);
    zero_f32_kernel<<<(int)((NF + 255) / 256), 256, 0, stream>>>(B, NF);
    scatter_kernel<<<scatBlocks, 256, 0, stream>>>(edges, A, B, E);
    zero_f32_kernel<<<1, 256, 0, stream>>>(stats, 256);
    bn_stats_kernel<<<(N + 63) / 64, 128, 0, stream>>>(B, ndst, b1, stats, N);
    bn_apply_kernel<<<(int)((NF + 255) / 256), 256, 0, stream>>>(
        B, ndst, b1, stats, g1, be1, a1, A, N);

    // ---- layer 2 ---------------------------------------------------------
    gemm_gcn_wmma<<<gemmGrid, 256, 0, stream>>>(A, nsrc, W2, B, N);
    zero_f32_kernel<<<(int)((NF + 255) / 256), 256, 0, stream>>>(A, NF);
    scatter_kernel<<<scatBlocks, 256, 0, stream>>>(edges, B, A, E);
    zero_f32_kernel<<<1, 256, 0, stream>>>(stats, 256);
    bn_stats_kernel<<<(N + 63) / 64, 128, 0, stream>>>(A, ndst, b2, stats, N);
    bn_apply_kernel<<<(int)((NF + 255) / 256), 256, 0, stream>>>(
        A, ndst, b2, stats, g2, be2, a2, A, N);   // in-place -> d_out
}